// Model_Loss_63969242906624
// MI455X (gfx1250) — compile-verified
//
#include <hip/hip_runtime.h>
#include <math.h>

typedef float v2f __attribute__((ext_vector_type(2)));
typedef float v8f __attribute__((ext_vector_type(8)));

#define NB 128
#define NN 64
#define NC 20
#define CAND (3 * NB * NN)          // 24576 candidates (level, b, n)
#define OFF1 819200                  // 128*80*80
#define OFF2 1024000                 // OFF1 + 128*40*40
#define CELLS 1075200                // OFF2 + 128*20*20  (multiple of 4)
#define CELLS4 (CELLS / 4)

// ---------------- math helpers ----------------
__device__ __forceinline__ float bce_logits(float x, float t) {
    // max(x,0) - x*t + log1p(exp(-|x|)); numerically stable
    return fmaxf(x, 0.0f) - x * t + log1pf(expf(-fabsf(x)));
}
__device__ __forceinline__ float smooth_l1(float p, float t) {
    float d = fabsf(p - t);
    return d < 1.0f ? 0.5f * d * d : d - 0.5f;
}

// ---------------- candidate decode ----------------
struct Cand {
    int valid, cell, lvl, b, gi, gj, dim;
    float cx, cy, w, h, stride;
};

__device__ __forceinline__ Cand decode_cand(int tid, const float* __restrict__ boxes) {
    Cand c;
    c.lvl  = tid / (NB * NN);
    int r  = tid % (NB * NN);            // r = b*NN + n, ravel order (last wins = max r)
    c.b    = r / NN;
    const float* bx = boxes + 4 * r;
    float x1 = bx[0], y1 = bx[1], x2 = bx[2], y2 = bx[3];
    c.cx = (x1 + x2) * 0.5f;
    c.cy = (y1 + y2) * 0.5f;
    c.w  = x2 - x1;
    c.h  = y2 - y1;
    c.stride = (c.lvl == 0) ? 8.0f : ((c.lvl == 1) ? 16.0f : 32.0f);
    c.dim    = (c.lvl == 0) ? 80   : ((c.lvl == 1) ? 40    : 20);
    c.gi = (int)floorf(c.cx / c.stride);
    c.gj = (int)floorf(c.cy / c.stride);
    c.valid = (c.gi >= 0) & (c.gi < c.dim) & (c.gj >= 0) & (c.gj < c.dim);
    int off = (c.lvl == 0) ? 0 : ((c.lvl == 1) ? OFF1 : OFF2);
    c.cell  = off + (c.b * c.dim + c.gj) * c.dim + c.gi;
    return c;
}

// ---------------- kernel 1: zero workspace (B128 stores) ----------------
// winner[CELLS] and cmask[CELLS] are adjacent -> one int4 stream zeroes both.
__global__ void k_init(int4* __restrict__ wz, float* __restrict__ acc) {
    int i  = blockIdx.x * blockDim.x + threadIdx.x;
    int st = gridDim.x * blockDim.x;
    const int n4 = (2 * CELLS) / 4;
    int4 z; z.x = 0; z.y = 0; z.z = 0; z.w = 0;
    for (int q = i; q < n4; q += st) wz[q] = z;
    if (i < 16) acc[i] = 0.0f;
}

// ---------------- kernel 2: dedupe scatter (winner key + class bitmask) --------
__global__ void k_scatter(const float* __restrict__ boxes, const int* __restrict__ labels,
                          int* __restrict__ winner, unsigned* __restrict__ cmask) {
    int tid = blockIdx.x * blockDim.x + threadIdx.x;
    if (tid >= CAND) return;
    Cand c = decode_cand(tid, boxes);
    if (!c.valid) return;
    int r = tid % (NB * NN);
    atomicMax(&winner[c.cell], r + 1);            // max r == last-write-wins
    atomicOr(&cmask[c.cell], 1u << labels[r]);    // union of labels in cell
}

// ---------------- kernel 3: sparse cls/box loss at positive cells ------------
__global__ void k_pos_loss(const float* __restrict__ boxes, const int* __restrict__ labels,
                           const float* __restrict__ cls0, const float* __restrict__ cls1,
                           const float* __restrict__ cls2,
                           const float* __restrict__ bx0, const float* __restrict__ bx1,
                           const float* __restrict__ bx2,
                           const int* __restrict__ winner, const unsigned* __restrict__ cmask,
                           float* __restrict__ acc) {
    int tid = blockIdx.x * blockDim.x + threadIdx.x;
    if (tid >= CAND) return;
    Cand c = decode_cand(tid, boxes);
    if (!c.valid) return;
    int r = tid % (NB * NN);
    if (winner[c.cell] != r + 1) return;          // only the winning candidate per cell

    int HW  = c.dim * c.dim;
    int off = (c.lvl == 0) ? 0 : ((c.lvl == 1) ? OFF1 : OFF2);
    int rem = (c.cell - off) - c.b * HW;          // gy*W + gx

    // cls loss over 20 channels vs. class-union indicator
    const float* cp = (c.lvl == 0) ? cls0 : ((c.lvl == 1) ? cls1 : cls2);
    unsigned m = cmask[c.cell];
    float cs = 0.0f;
#pragma unroll
    for (int ch = 0; ch < NC; ++ch) {
        float x = cp[(c.b * NC + ch) * HW + rem];
        float t = ((m >> ch) & 1u) ? 1.0f : 0.0f;
        cs += bce_logits(x, t);
    }

    // box loss: winner's tvals are exactly the scattered targets
    float invs = 1.0f / c.stride;
    float tv[4];
    tv[0] = c.cx * invs - (float)c.gi;
    tv[1] = c.cy * invs - (float)c.gj;
    tv[2] = logf(c.w * invs + 1e-6f);
    tv[3] = logf(c.h * invs + 1e-6f);
    const float* bp = (c.lvl == 0) ? bx0 : ((c.lvl == 1) ? bx1 : bx2);
    float bs = 0.0f;
#pragma unroll
    for (int j = 0; j < 4; ++j)
        bs += smooth_l1(bp[(c.b * 4 + j) * HW + rem], tv[j]);

    atomicAdd(&acc[1 + c.lvl], cs);   // per-level cls sum
    atomicAdd(&acc[4 + c.lvl], bs);   // per-level box sum
    atomicAdd(&acc[7 + c.lvl], 1.0f); // per-level num_pos
}

// ---------------- kernel 4: dense obj BCE pass, B128 loads + WMMA reduce -----
// OFF1/OFF2/CELLS are multiples of 4, so each 4-cell chunk is within one level.
__global__ void k_obj_loss(const float* __restrict__ obj0, const float* __restrict__ obj1,
                           const float* __restrict__ obj2, const int* __restrict__ winner,
                           float* __restrict__ acc) {
    int tid = blockIdx.x * blockDim.x + threadIdx.x;
    int st  = gridDim.x * blockDim.x;
    const int4* __restrict__ w4p = (const int4*)winner;
    v2f a;  a.x = 0.0f; a.y = 0.0f;   // two per-lane partials -> 64 values per wave

    for (int q = tid; q < CELLS4; q += st) {
        int ci  = q * 4;
        int lvl = (ci >= OFF2) ? 2 : ((ci >= OFF1) ? 1 : 0);
        int off = (lvl == 0) ? 0 : ((lvl == 1) ? OFF1 : OFF2);
        const float4* __restrict__ p =
            (const float4*)((lvl == 0) ? obj0 : ((lvl == 1) ? obj1 : obj2));
        float wgt = (lvl == 0) ? (1.0f / 819200.0f)
                  : (lvl == 1) ? (1.0f / 204800.0f) : (1.0f / 51200.0f);

        float4 x = p[(ci - off) >> 2];             // global_load_b128
        int4   w = w4p[q];                          // global_load_b128

        float v0 = bce_logits(x.x, w.x ? 1.0f : 0.0f);
        float v1 = bce_logits(x.y, w.y ? 1.0f : 0.0f);
        float v2 = bce_logits(x.z, w.z ? 1.0f : 0.0f);
        float v3 = bce_logits(x.w, w.w ? 1.0f : 0.0f);
        a.x += (v0 + v1) * wgt;
        a.y += (v2 + v3) * wgt;
    }

    // WMMA wave reduction: A = 16x4 of partials, B = ones(4x16)
    // => D[m][n] = rowsum_m(A). Lanes 0-15 carry rows 0-7 in d[0..7],
    //    lanes 16-31 rows 8-15, so total = s + shfl_xor(s, 16).
    v2f ones; ones.x = 1.0f; ones.y = 1.0f;
    v8f cz = {};
    v8f d = __builtin_amdgcn_wmma_f32_16x16x4_f32(false, a, false, ones,
                                                  (short)0, cz, false, false);
    float s = d[0] + d[1] + d[2] + d[3] + d[4] + d[5] + d[6] + d[7];
    s += __shfl_xor(s, 16, 32);
    if ((threadIdx.x & 31) == 0) atomicAdd(&acc[0], s);
}

// ---------------- kernel 5: finalize ----------------
__global__ void k_finalize(const float* __restrict__ acc, float* __restrict__ out) {
    if (blockIdx.x == 0 && threadIdx.x == 0) {
        float lc = 0.0f, lb = 0.0f;
#pragma unroll
        for (int l = 0; l < 3; ++l) {
            float np = fmaxf(acc[7 + l], 1.0f);
            lc += acc[1 + l] / np;
            lb += acc[4 + l] / np;
        }
        float lo = acc[0];
        out[0] = lc + lb + lo;  // total
        out[1] = lc;            // loss_cls
        out[2] = lb;            // loss_box
        out[3] = lo;            // loss_obj
    }
}

extern "C" void kernel_launch(void* const* d_in, const int* in_sizes, int n_in,
                              void* d_out, int out_size, void* d_ws, size_t ws_size,
                              hipStream_t stream) {
    // setup_inputs() dict order: cls0, box0, obj0, cls1, box1, obj1, cls2, box2, obj2, boxes, labels
    const float* cls0  = (const float*)d_in[0];
    const float* box0  = (const float*)d_in[1];
    const float* obj0  = (const float*)d_in[2];
    const float* cls1  = (const float*)d_in[3];
    const float* box1  = (const float*)d_in[4];
    const float* obj1  = (const float*)d_in[5];
    const float* cls2  = (const float*)d_in[6];
    const float* box2  = (const float*)d_in[7];
    const float* obj2  = (const float*)d_in[8];
    const float* boxes = (const float*)d_in[9];
    const int*   labels= (const int*)d_in[10];
    float* out = (float*)d_out;

    // workspace: winner[CELLS] int | cmask[CELLS] u32 | acc[16] f32  (~8.6 MB)
    int*      winner = (int*)d_ws;
    unsigned* cmask  = (unsigned*)d_ws + CELLS;
    float*    acc    = (float*)d_ws + 2 * (size_t)CELLS;

    k_init<<<1024, 256, 0, stream>>>((int4*)d_ws, acc);
    k_scatter<<<(CAND + 255) / 256, 256, 0, stream>>>(boxes, labels, winner, cmask);
    k_pos_loss<<<(CAND + 255) / 256, 256, 0, stream>>>(boxes, labels,
                                                       cls0, cls1, cls2,
                                                       box0, box1, box2,
                                                       winner, cmask, acc);
    k_obj_loss<<<512, 256, 0, stream>>>(obj0, obj1, obj2, winner, acc);
    k_finalize<<<1, 32, 0, stream>>>(acc, out);
}